// MultiHeadSelfAttention_34763465294677
// MI455X (gfx1250) — compile-verified
//
#include <hip/hip_runtime.h>

typedef float v2f __attribute__((ext_vector_type(2)));
typedef float v8f __attribute__((ext_vector_type(8)));

namespace {
constexpr int BB = 4;     // batch
constexpr int T  = 2048;  // sequence
constexpr int E  = 2048;  // embed
constexpr int NH = 16;    // heads
constexpr int HD = 128;   // head dim
}

static __device__ __forceinline__ v8f wmma4(v2f a, v2f b, v8f c) {
  // V_WMMA_F32_16X16X4_F32 : D = A(16x4) * B(4x16) + C(16x16), all f32
  return __builtin_amdgcn_wmma_f32_16x16x4_f32(false, a, false, b, (short)0, c,
                                               false, false);
}

static __device__ __forceinline__ v8f v8f_zero() {
  v8f z;
#pragma unroll
  for (int e = 0; e < 8; ++e) z[e] = 0.0f;
  return z;
}

// A-fragment (and mirrored B-fragment) loader for 16x16x4 f32 WMMA:
// lane L reads row (lane&15), K-pair (2*(lane>>4)) .. +1
static __device__ __forceinline__ v2f ldfrag(const float* s, int stride,
                                             int row0, int kk, int lane) {
  const float* p = s + (row0 + (lane & 15)) * stride + kk + ((lane >> 4) << 1);
  v2f f;
  f.x = p[0];
  f.y = p[1];
  return f;
}

// ---------------------------------------------------------------------------
// Kernel 1: y = x @ W^T for W in {Wq,Wk,Wv} (blockIdx.z selects).
// q -> workspace [B,H,T,hd]; k,v -> output segments [B,H,T,hd].
// ---------------------------------------------------------------------------
__global__ __launch_bounds__(256) void mhsa_qkv_gemm(
    const float* __restrict__ x, const float* __restrict__ Wq,
    const float* __restrict__ Wk, const float* __restrict__ Wv,
    float* __restrict__ qws, float* __restrict__ kout,
    float* __restrict__ vout) {
  constexpr int SA = 20;  // padded LDS row stride (16 + 4): aligned + no bank conflicts
  __shared__ float As[128 * SA];
  __shared__ float Bs[128 * SA];
  const int tid = threadIdx.x;
  const int lane = tid & 31;
  const int wave = tid >> 5;
  const int wm = wave >> 1;  // 0..3 : wave row
  const int wn = wave & 1;   // 0..1 : wave col
  const int z = blockIdx.z;
  const float* __restrict__ W = (z == 0) ? Wq : (z == 1) ? Wk : Wv;
  const int m0 = blockIdx.y * 128;
  const int n0 = blockIdx.x * 128;

  v8f acc[2][4];
#pragma unroll
  for (int i = 0; i < 2; ++i)
#pragma unroll
    for (int j = 0; j < 4; ++j) acc[i][j] = v8f_zero();

  for (int k0 = 0; k0 < E; k0 += 16) {
#pragma unroll
    for (int it = 0; it < 2; ++it) {
      const int idx = tid * 4 + it * 1024;  // over 128x16 tile
      const int r = idx >> 4;
      const int c = idx & 15;
      *(float4*)(As + r * SA + c) =
          *(const float4*)(x + (size_t)(m0 + r) * E + k0 + c);
      *(float4*)(Bs + r * SA + c) =
          *(const float4*)(W + (size_t)(n0 + r) * E + k0 + c);
    }
    __syncthreads();
#pragma unroll
    for (int kk = 0; kk < 16; kk += 4) {
      v2f a[2], b[4];
#pragma unroll
      for (int i = 0; i < 2; ++i) a[i] = ldfrag(As, SA, wm * 32 + i * 16, kk, lane);
#pragma unroll
      for (int j = 0; j < 4; ++j) b[j] = ldfrag(Bs, SA, wn * 64 + j * 16, kk, lane);
#pragma unroll
      for (int i = 0; i < 2; ++i)
#pragma unroll
        for (int j = 0; j < 4; ++j) acc[i][j] = wmma4(a[i], b[j], acc[i][j]);
    }
    __syncthreads();
  }

  float* __restrict__ dst = (z == 0) ? qws : (z == 1) ? kout : vout;
  const int hp8 = (lane >> 4) << 3;
#pragma unroll
  for (int i = 0; i < 2; ++i)
#pragma unroll
    for (int j = 0; j < 4; ++j) {
      const int gcol = n0 + wn * 64 + j * 16 + (lane & 15);  // e = h*HD + d
      const int hh = gcol >> 7;
      const int dd = gcol & (HD - 1);
#pragma unroll
      for (int r = 0; r < 8; ++r) {
        const int grow = m0 + wm * 32 + i * 16 + r + hp8;  // bt = b*T + t
        const int bb = grow >> 11;
        const int tt = grow & (T - 1);
        dst[(((size_t)bb * NH + hh) * T + tt) * HD + dd] = acc[i][j][r];
      }
    }
}

// ---------------------------------------------------------------------------
// Kernel 2: causal flash attention, f32 WMMA. One block = (b, h, 64 q rows),
// 4 waves x 16 q rows. Computes S^T = K Q^T then O^T += V^T P^T so the
// softmax state lives on lanes (q = lane&15). Writes O^T as [B, E, T].
// ---------------------------------------------------------------------------
__global__ __launch_bounds__(128) void mhsa_flash_attn(
    const float* __restrict__ q, const float* __restrict__ k,
    const float* __restrict__ v, float* __restrict__ oT) {
  constexpr int QS = 132;  // padded strides
  constexpr int KS = 132;
  __shared__ float Qs[64 * QS];
  __shared__ float Ks[16 * KS];
  __shared__ float Vs[16 * KS];
  __shared__ float Pt[4][16 * 17];

  const int tid = threadIdx.x;
  const int lane = tid & 31;
  const int wave = tid >> 5;
  const int b = blockIdx.z;
  const int h = blockIdx.y;
  const int q0 = blockIdx.x * 64;
  const int wq0 = q0 + wave * 16;
  const size_t bh = ((size_t)b * NH + h) * T * HD;
  const float scale = 0.088388347648318447f;  // 1/sqrt(128)
  const int h2 = (lane >> 4) << 1;

  // Stage the 64x128 Q block into LDS (coalesced), then hoist Q^T fragments.
#pragma unroll
  for (int it = 0; it < 16; ++it) {
    const int idx = tid * 4 + it * 512;
    const int r = idx >> 7;
    const int c = idx & 127;
    *(float4*)(Qs + r * QS + c) =
        *(const float4*)(q + bh + (size_t)(q0 + r) * HD + c);
  }
  __syncthreads();

  v2f qf[32];
  const int qrow = wave * 16 + (lane & 15);
#pragma unroll
  for (int ds = 0; ds < 32; ++ds) {
    const float* p = Qs + qrow * QS + ds * 4 + h2;
    qf[ds].x = p[0] * scale;
    qf[ds].y = p[1] * scale;
  }

  v8f o[8];
#pragma unroll
  for (int dt = 0; dt < 8; ++dt) o[dt] = v8f_zero();
  float m = -__builtin_inff();
  float l = 0.0f;

  const int nblk = (q0 >> 4) + 4;  // kv tiles up to this block's diagonal
  for (int kb = 0; kb < nblk; ++kb) {
    __syncthreads();  // previous iteration's consumers are done with Ks/Vs
    const int kv0 = kb << 4;
#pragma unroll
    for (int it = 0; it < 4; ++it) {
      const int idx = tid * 4 + it * 512;
      const int r = idx >> 7;
      const int c = idx & 127;
      *(float4*)(Ks + r * KS + c) =
          *(const float4*)(k + bh + (size_t)(kv0 + r) * HD + c);
      *(float4*)(Vs + r * KS + c) =
          *(const float4*)(v + bh + (size_t)(kv0 + r) * HD + c);
    }
    __syncthreads();

    if (kv0 <= wq0 + 15) {  // wave-uniform: EXEC stays all-ones for WMMA
      // S^T (16x16) = K_tile(16xHD) * Q_strip^T(HDx16), accumulate over d
      v8f st = v8f_zero();
#pragma unroll
      for (int ds = 0; ds < 32; ++ds) {
        v2f kf;
        const float* p = Ks + (lane & 15) * KS + ds * 4 + h2;
        kf.x = p[0];
        kf.y = p[1];
        st = wmma4(kf, qf[ds], st);
      }
      // causal mask on the diagonal block (S^T layout: row j on vgprs, q on lanes)
      if (kv0 + 15 > wq0) {
        const int qg = wq0 + (lane & 15);
        const int jb = kv0 + ((lane >> 4) << 3);
#pragma unroll
        for (int r = 0; r < 8; ++r)
          if (jb + r > qg) st[r] = -__builtin_inff();
      }
      // online softmax: per-lane state for q = lane&15 (mirrored in both halves)
      float mb = st[0];
#pragma unroll
      for (int r = 1; r < 8; ++r) mb = fmaxf(mb, st[r]);
      mb = fmaxf(mb, __shfl_xor(mb, 16, 32));
      const float mn = fmaxf(m, mb);
      const float alpha = __expf(m - mn);
      float ls = 0.0f;
#pragma unroll
      for (int r = 0; r < 8; ++r) {
        st[r] = __expf(st[r] - mn);
        ls += st[r];
      }
      ls += __shfl_xor(ls, 16, 32);
      l = l * alpha + ls;
      m = mn;
#pragma unroll
      for (int dt = 0; dt < 8; ++dt)
#pragma unroll
        for (int r = 0; r < 8; ++r) o[dt][r] *= alpha;

      // P^T roundtrip through per-wave LDS to reach WMMA operand layout
      float* pt = &Pt[wave][0];
      const int jb8 = (lane >> 4) << 3;
#pragma unroll
      for (int r = 0; r < 8; ++r) pt[(jb8 + r) * 17 + (lane & 15)] = st[r];
      asm volatile("s_wait_dscnt 0" ::: "memory");  // wave-local DS RAW fence

      // O^T(HDx16) += V^T * P^T : A = V^T fragments, B = P^T fragments
#pragma unroll
      for (int js = 0; js < 4; ++js) {
        const int j = js * 4 + h2;
        v2f pf;
        pf.x = pt[j * 17 + (lane & 15)];
        pf.y = pt[(j + 1) * 17 + (lane & 15)];
#pragma unroll
        for (int dt = 0; dt < 8; ++dt) {
          v2f vf;
          vf.x = Vs[j * KS + dt * 16 + (lane & 15)];
          vf.y = Vs[(j + 1) * KS + dt * 16 + (lane & 15)];
          o[dt] = wmma4(vf, pf, o[dt]);
        }
      }
    }
  }

  // Store O^T as [B, E, T]: lanes run over q -> coalesced columns
  const float invl = 1.0f / l;
  const int qq = wq0 + (lane & 15);
  const int hp8 = (lane >> 4) << 3;
#pragma unroll
  for (int dt = 0; dt < 8; ++dt)
#pragma unroll
    for (int r = 0; r < 8; ++r) {
      const int d = dt * 16 + r + hp8;
      oT[((size_t)b * E + h * HD + d) * T + qq] = o[dt][r] * invl;
    }
}

// ---------------------------------------------------------------------------
// Kernel 3: out = attn @ Wp^T + bp, reading attn^T stored as [B, E, T].
// ---------------------------------------------------------------------------
__global__ __launch_bounds__(256) void mhsa_proj_gemm(
    const float* __restrict__ aT, const float* __restrict__ Wp,
    const float* __restrict__ bp, float* __restrict__ out) {
  constexpr int SA = 20;
  __shared__ float As[128 * SA];
  __shared__ float Bs[128 * SA];
  const int tid = threadIdx.x;
  const int lane = tid & 31;
  const int wave = tid >> 5;
  const int wm = wave >> 1;
  const int wn = wave & 1;
  const int m0 = blockIdx.y * 128;
  const int n0 = blockIdx.x * 128;
  const int b = m0 >> 11;       // 128-row tiles never cross a batch boundary
  const int t0 = m0 & (T - 1);

  v8f acc[2][4];
#pragma unroll
  for (int i = 0; i < 2; ++i)
#pragma unroll
    for (int j = 0; j < 4; ++j) acc[i][j] = v8f_zero();

  for (int k0 = 0; k0 < E; k0 += 16) {
#pragma unroll
    for (int it = 0; it < 2; ++it) {
      const int idx = tid * 4 + it * 1024;
      // A: read [e][t] rows coalesced, write transposed [t][e] into LDS
      const int c = idx >> 7;
      const int tt = idx & 127;
      const float4 a4 =
          *(const float4*)(aT + ((size_t)b * E + k0 + c) * T + t0 + tt);
      As[(tt + 0) * SA + c] = a4.x;
      As[(tt + 1) * SA + c] = a4.y;
      As[(tt + 2) * SA + c] = a4.z;
      As[(tt + 3) * SA + c] = a4.w;
      // B: Wp rows, same as QKV kernel
      const int r = idx >> 4;
      const int cc = idx & 15;
      *(float4*)(Bs + r * SA + cc) =
          *(const float4*)(Wp + (size_t)(n0 + r) * E + k0 + cc);
    }
    __syncthreads();
#pragma unroll
    for (int kk = 0; kk < 16; kk += 4) {
      v2f a[2], bfr[4];
#pragma unroll
      for (int i = 0; i < 2; ++i) a[i] = ldfrag(As, SA, wm * 32 + i * 16, kk, lane);
#pragma unroll
      for (int j = 0; j < 4; ++j) bfr[j] = ldfrag(Bs, SA, wn * 64 + j * 16, kk, lane);
#pragma unroll
      for (int i = 0; i < 2; ++i)
#pragma unroll
        for (int j = 0; j < 4; ++j) acc[i][j] = wmma4(a[i], bfr[j], acc[i][j]);
    }
    __syncthreads();
  }

  const int hp8 = (lane >> 4) << 3;
#pragma unroll
  for (int i = 0; i < 2; ++i)
#pragma unroll
    for (int j = 0; j < 4; ++j) {
      const int gcol = n0 + wn * 64 + j * 16 + (lane & 15);
      const float bias = bp[gcol];
#pragma unroll
      for (int r = 0; r < 8; ++r) {
        const int grow = m0 + wm * 32 + i * 16 + r + hp8;
        out[(size_t)grow * E + gcol] = acc[i][j][r] + bias;
      }
    }
}

extern "C" void kernel_launch(void* const* d_in, const int* in_sizes, int n_in,
                              void* d_out, int out_size, void* d_ws,
                              size_t ws_size, hipStream_t stream) {
  const float* x  = (const float*)d_in[0];
  const float* Wq = (const float*)d_in[1];
  const float* Wk = (const float*)d_in[2];
  const float* Wv = (const float*)d_in[3];
  const float* Wp = (const float*)d_in[4];
  const float* bp = (const float*)d_in[5];

  const size_t seg = (size_t)BB * NH * T * HD;  // 16,777,216 floats
  float* out  = (float*)d_out;   // [B,T,E]
  float* kout = out + seg;       // [B,H,T,hd]
  float* vout = kout + seg;      // [B,H,T,hd]

  // Workspace: q [B,H,T,hd] (64MB) + attn^T [B,E,T] (64MB) = 128MB
  float* qws   = (float*)d_ws;
  float* attnT = qws + seg;

  mhsa_qkv_gemm<<<dim3(E / 128, (BB * T) / 128, 3), 256, 0, stream>>>(
      x, Wq, Wk, Wv, qws, kout, vout);
  mhsa_flash_attn<<<dim3(T / 64, NH, BB), 128, 0, stream>>>(qws, kout, vout,
                                                            attnT);
  mhsa_proj_gemm<<<dim3(E / 128, (BB * T) / 128), 256, 0, stream>>>(attnT, Wp,
                                                                    bp, out);
}